// GraphTotalVariation_35588099014976
// MI455X (gfx1250) — compile-verified
//
#include <hip/hip_runtime.h>
#include <math.h>

// GraphTotalVariation loss for MI455X (gfx1250), wave32.
// d2 tile computed fully inside V_WMMA_F32_16X16X4_F32:
//   A (cols, LDS)  = (-2x_j, -2y_j, -2z_j, sq_j)   (16x4)
//   B (rows, regs) = (  x_i,   y_i,   z_i, 1.0 )   (4x16)
//   C              = sq_i broadcast
//   D[m][n]        = sq_i + sq_j - 2*dot = squared distance
// All 32 lanes keep an independent top-16 (lane pair (l, l+16) covers
// disjoint column halves of row l); lists merged through LDS at the end.

typedef __attribute__((ext_vector_type(2))) float v2f;
typedef __attribute__((ext_vector_type(8))) float v8f;

#define NPB_        8192
#define BATCHES_    4
#define KNN_        16
#define ROWS_PER_BLOCK_ 128     // 8 waves x 16 rows
#define COL_TILE_   2048        // 32 KB LDS tile
#define THREADS_    256
#define LIST_STRIDE_ (KNN_ + 1) // padded to avoid LDS bank conflicts in merge

__global__ __launch_bounds__(THREADS_)
void graph_tv_loss_kernel(const float* __restrict__ coord,
                          const float* __restrict__ outv,
                          float* __restrict__ loss)
{
    // first 8192 floats: column tile (-2x,-2y,-2z,sq); reused at the end as
    // per-lane top-K lists (256 lanes * 17 floats + 256 lanes * 17 ints)
    __shared__ float sMem[2 * THREADS_ * LIST_STRIDE_];   // 8704 floats (34 KB)

    const int tid  = threadIdx.x;
    const int lane = tid & 31;
    const int wave = tid >> 5;        // 0..7
    const int half = lane >> 4;       // 0: K0/K1 half, 1: K2/K3 half
    const int l16  = lane & 15;

    const int blocksPerBatch = NPB_ / ROWS_PER_BLOCK_;     // 64
    const int b        = blockIdx.x / blocksPerBatch;
    const int rowBase  = (blockIdx.x % blocksPerBatch) * ROWS_PER_BLOCK_ + wave * 16;

    // ---------------- row (i) setup ----------------
    const int rowLocal = rowBase + l16;          // 0..8191 within batch
    const int ip       = b * NPB_ + rowLocal;    // global point index
    const float4* coordq = (const float4*)coord;
    const float4 c4    = coordq[ip];             // (batch_id, x, y, z)
    const float xi = c4.y, yi = c4.z, zi = c4.w;
    const float sqi = xi * xi + yi * yi + zi * zi;

    v2f bmat;                                    // B operand (4x16, K x N)
    if (half == 0) { bmat[0] = xi; bmat[1] = yi; }
    else           { bmat[0] = zi; bmat[1] = 1.0f; }

    v8f cmat;                                    // C = sq_i broadcast
    #pragma unroll
    for (int t = 0; t < 8; ++t) cmat[t] = sqi;

    // ---------------- per-lane top-K state (sorted ascending) ----------------
    float bd[KNN_];
    int   bi[KNN_];
    #pragma unroll
    for (int t = 0; t < KNN_; ++t) { bd[t] = 3.0e38f; bi[t] = 0; }

    // branchless swap-chain insert under a single divergence guard
    auto processTile = [&](v8f d, int base) {
        #pragma unroll
        for (int v = 0; v < 8; ++v) {
            float c = fmaxf(d[v], 0.0f);
            int   j = base + v;
            if (j != rowLocal && c < bd[KNN_ - 1]) {
                float cd = c; int ci = j;
                #pragma unroll
                for (int t = 0; t < KNN_; ++t) {
                    bool  lt   = cd < bd[t];
                    float vmin = lt ? cd    : bd[t];
                    float vmax = lt ? bd[t] : cd;
                    int   imin = lt ? ci    : bi[t];
                    int   imax = lt ? bi[t] : ci;
                    bd[t] = vmin; cd = vmax;
                    bi[t] = imin; ci = imax;
                }
            }
        }
    };

    const v2f* sCol2 = (const v2f*)sMem;

    // ---------------- scan all columns of this batch ----------------
    for (int t0 = 0; t0 < NPB_; t0 += COL_TILE_) {
        // cooperative LDS stage of 2048 column points
        for (int e = tid; e < COL_TILE_; e += THREADS_) {
            float4 p = coordq[b * NPB_ + t0 + e];
            float sq = p.y * p.y + p.z * p.z + p.w * p.w;
            sMem[e * 4 + 0] = -2.0f * p.y;
            sMem[e * 4 + 1] = -2.0f * p.z;
            sMem[e * 4 + 2] = -2.0f * p.w;
            sMem[e * 4 + 3] = sq;
        }
        __syncthreads();

        // prefetch next tile's coords into cache while we compute
        if (t0 + COL_TILE_ < NPB_)
            __builtin_prefetch(coordq + b * NPB_ + t0 + COL_TILE_ + tid * 8, 0, 1);

        for (int cb = 0; cb < COL_TILE_; cb += 64) {
            // batch 4 A-operand loads, then 4 independent WMMAs
            v2f a0 = sCol2[(cb +  0 + l16) * 2 + half];
            v2f a1 = sCol2[(cb + 16 + l16) * 2 + half];
            v2f a2 = sCol2[(cb + 32 + l16) * 2 + half];
            v2f a3 = sCol2[(cb + 48 + l16) * 2 + half];

            v8f d0 = __builtin_amdgcn_wmma_f32_16x16x4_f32(
                false, a0, false, bmat, (short)0, cmat, false, false);
            v8f d1 = __builtin_amdgcn_wmma_f32_16x16x4_f32(
                false, a1, false, bmat, (short)0, cmat, false, false);
            v8f d2 = __builtin_amdgcn_wmma_f32_16x16x4_f32(
                false, a2, false, bmat, (short)0, cmat, false, false);
            v8f d3 = __builtin_amdgcn_wmma_f32_16x16x4_f32(
                false, a3, false, bmat, (short)0, cmat, false, false);

            // lane's own columns: base + {0..7}, where base depends on lane half
            const int cBase = t0 + cb + half * 8;
            processTile(d0, cBase);
            processTile(d1, cBase + 16);
            processTile(d2, cBase + 32);
            processTile(d3, cBase + 48);
        }
        __syncthreads();
    }

    // ---------------- dump per-lane lists to LDS, merge lane pairs ----------------
    float* sD = sMem;                                     // 256*17 floats
    int*   sI = (int*)(sMem + THREADS_ * LIST_STRIDE_);   // 256*17 ints
    {
        const int base = tid * LIST_STRIDE_;
        #pragma unroll
        for (int k = 0; k < KNN_; ++k) { sD[base + k] = bd[k]; sI[base + k] = bi[k]; }
    }
    __syncthreads();

    float lsum = 0.0f;
    if (half == 0) {
        // normalize out_i
        const float* oi = outv + (size_t)ip * 3;
        float ox = oi[0], oy = oi[1], oz = oi[2];
        float ni = sqrtf(ox * ox + oy * oy + oz * oz);
        float inv = 1.0f / fmaxf(ni, 1e-12f);
        ox *= inv; oy *= inv; oz *= inv;

        // streaming 2-way merge of the lane pair's sorted lists; consume the
        // 16 smallest directly into the loss (no merged array needed)
        const int baseA = tid * LIST_STRIDE_;
        const int baseB = (tid + 16) * LIST_STRIDE_;
        int pa = 0, pb = 0;
        #pragma unroll
        for (int k = 0; k < KNN_; ++k) {
            float da = sD[baseA + pa];
            float db = sD[baseB + pb];
            bool  ta = da <= db;
            float dk = ta ? da : db;
            int   jk = ta ? sI[baseA + pa] : sI[baseB + pb];
            pa += ta ? 1 : 0;
            pb += ta ? 0 : 1;

            const float* oj = outv + (size_t)(b * NPB_ + jk) * 3;
            float jx = oj[0], jy = oj[1], jz = oj[2];
            float nj = sqrtf(jx * jx + jy * jy + jz * jz);
            float invj = 1.0f / fmaxf(nj, 1e-12f);
            jx *= invj; jy *= invj; jz *= invj;
            float w = __expf(-dk * 0.25f);               // exp(-d2 / gamma^2), gamma=2
            lsum += w * (fabsf(ox - jx) + fabsf(oy - jy) + fabsf(oz - jz));
        }
    }

    // wave32 reduction, then one atomic per wave
    #pragma unroll
    for (int off = 16; off > 0; off >>= 1)
        lsum += __shfl_xor(lsum, off, 32);
    if (lane == 0)
        atomicAdd(loss, lsum * (1.0f / (float)(BATCHES_ * NPB_ * KNN_)));
}

extern "C" void kernel_launch(void* const* d_in, const int* in_sizes, int n_in,
                              void* d_out, int out_size, void* d_ws, size_t ws_size,
                              hipStream_t stream) {
    const float* coord = (const float*)d_in[0];   // (N,4): [batch, x, y, z]
    // d_in[1] = intensity (unused)
    const float* outv  = (const float*)d_in[2];   // (N,3)
    // d_in[3] = target (unused)
    float* loss = (float*)d_out;                  // scalar f32

    hipMemsetAsync(loss, 0, sizeof(float), stream);

    dim3 grid(BATCHES_ * (NPB_ / ROWS_PER_BLOCK_));   // 256 blocks
    dim3 block(THREADS_);                              // 8 waves
    graph_tv_loss_kernel<<<grid, block, 0, stream>>>(coord, outv, loss);
}